// StaticGraphConvolution_85727547228210
// MI455X (gfx1250) — compile-verified
//
#include <hip/hip_runtime.h>
#include <hip/hip_bf16.h>
#include <stdint.h>

// ---------------- problem constants (from reference) ----------------
#define D_FEAT            64
#define ALPHA             0.1f
#define ONE_MINUS_ALPHA   0.9f

// ---------------- tuning ----------------
#define EPW               128   // edges per wave (E=1.6M -> 12500 waves)
#define WAVES_PER_BLOCK   8
#define BLOCK_THREADS     (WAVES_PER_BLOCK * 32)

#ifndef __has_builtin
#define __has_builtin(x) 0
#endif

// Async-load builtin existence is proven (round-1 diagnostic showed its
// parameter types). Gate only on it; handle the wait separately so a missing
// wait builtin cannot silently disable the async path.
#if defined(__AMDGCN__) && __has_builtin(__builtin_amdgcn_global_load_async_to_lds_b128)
#define USE_ASYNC_LDS 1
#else
#define USE_ASYNC_LDS 0
#endif

typedef int sgc_v4i __attribute__((vector_size(16)));

__device__ __forceinline__ void async_copy_b128_to_lds(const void* g, void* l) {
#if USE_ASYNC_LDS
  __builtin_amdgcn_global_load_async_to_lds_b128(
      (__attribute__((address_space(1))) sgc_v4i*)g,
      (__attribute__((address_space(3))) sgc_v4i*)l,
      /*offset=*/0, /*cpol=*/0);
#else
  *(float4*)l = *(const float4*)g;
#endif
}

__device__ __forceinline__ void async_wait_all() {
#if USE_ASYNC_LDS
#if __has_builtin(__builtin_amdgcn_s_wait_asynccnt)
  __builtin_amdgcn_s_wait_asynccnt(0);
#else
  asm volatile("s_wait_asynccnt 0x0" ::: "memory");
#endif
#endif
}

// ---------------- pass 1: out = ALPHA * features0 ----------------
__global__ __launch_bounds__(256) void sgc_init_kernel(const float* __restrict__ f0,
                                                       float* __restrict__ out, int n4) {
  int i = blockIdx.x * blockDim.x + threadIdx.x;
  if (i < n4) {
    float4 v = ((const float4*)f0)[i];
    v.x *= ALPHA; v.y *= ALPHA; v.z *= ALPHA; v.w *= ALPHA;
    ((float4*)out)[i] = v;
  }
}

// ---------------- pass 2: segmented SpMM scatter ----------------
// One wave handles EPW contiguous (sorted-by-row) edges.
// Lane L owns feature dims [2L, 2L+1]. Row changes flush via atomicAdd.
__global__ __launch_bounds__(BLOCK_THREADS) void sgc_scatter_kernel(
    const int*   __restrict__ rows,
    const int*   __restrict__ cols,
    const float* __restrict__ vals,
    const float* __restrict__ feat,
    float*       __restrict__ out,
    int nEdges, int nNodes) {
  __shared__ __align__(16) int   sRow[WAVES_PER_BLOCK][EPW];
  __shared__ __align__(16) int   sCol[WAVES_PER_BLOCK][EPW];
  __shared__ __align__(16) float sVal[WAVES_PER_BLOCK][EPW];

  const int wave = threadIdx.x >> 5;
  const int lane = threadIdx.x & 31;
  const long e0  = ((long)blockIdx.x * WAVES_PER_BLOCK + wave) * EPW;
  if (e0 >= nEdges) return;  // wave-uniform exit
  const int cnt = (nEdges - e0 < EPW) ? (int)(nEdges - e0) : EPW;

  if (cnt == EPW) {
    // Stage the whole chunk: each lane async-copies 4 dwords of each array
    // (3 x global_load_async_to_lds_b128 per wave, tracked by ASYNCcnt).
    const int o = lane * 4;
    async_copy_b128_to_lds(rows + e0 + o, &sRow[wave][o]);
    async_copy_b128_to_lds(cols + e0 + o, &sCol[wave][o]);
    async_copy_b128_to_lds(vals + e0 + o, &sVal[wave][o]);
    async_wait_all();
  } else {
    for (int j = lane; j < cnt; j += 32) {
      sRow[wave][j] = rows[e0 + j];
      sCol[wave][j] = cols[e0 + j];
      sVal[wave][j] = vals[e0 + j];
    }
  }
  // Per-wave producer/consumer only: no cross-wave barrier needed.

  const float2* __restrict__ f2 = (const float2*)feat;
  float ax = 0.f, ay = 0.f;
  int cur = sRow[wave][0];

#define SGC_FLUSH_IF_NEW(R)                                            \
  do {                                                                 \
    if ((R) != cur) {                                                  \
      if ((unsigned)cur < (unsigned)nNodes) {                          \
        float* dst = out + (size_t)cur * D_FEAT + (lane << 1);         \
        atomicAdd(dst + 0, ONE_MINUS_ALPHA * ax);                      \
        atomicAdd(dst + 1, ONE_MINUS_ALPHA * ay);                      \
      }                                                                \
      ax = 0.f; ay = 0.f; cur = (R);                                   \
    }                                                                  \
  } while (0)

  int j = 0;
  for (; j + 4 <= cnt; j += 4) {
    const int r0 = sRow[wave][j + 0], r1 = sRow[wave][j + 1];
    const int r2 = sRow[wave][j + 2], r3 = sRow[wave][j + 3];
    const int c0 = sCol[wave][j + 0], c1 = sCol[wave][j + 1];
    const int c2 = sCol[wave][j + 2], c3 = sCol[wave][j + 3];
    const float v0 = sVal[wave][j + 0], v1 = sVal[wave][j + 1];
    const float v2 = sVal[wave][j + 2], v3 = sVal[wave][j + 3];
    // Issue 4 independent gathers (each is one coalesced 256B row slice).
    const float2 g0 = f2[c0 * 32 + lane];
    const float2 g1 = f2[c1 * 32 + lane];
    const float2 g2 = f2[c2 * 32 + lane];
    const float2 g3 = f2[c3 * 32 + lane];

    SGC_FLUSH_IF_NEW(r0);
    ax = fmaf(v0, g0.x, ax); ay = fmaf(v0, g0.y, ay);
    SGC_FLUSH_IF_NEW(r1);
    ax = fmaf(v1, g1.x, ax); ay = fmaf(v1, g1.y, ay);
    SGC_FLUSH_IF_NEW(r2);
    ax = fmaf(v2, g2.x, ax); ay = fmaf(v2, g2.y, ay);
    SGC_FLUSH_IF_NEW(r3);
    ax = fmaf(v3, g3.x, ax); ay = fmaf(v3, g3.y, ay);
  }
  for (; j < cnt; ++j) {
    const int   r = sRow[wave][j];
    const int   c = sCol[wave][j];
    const float v = sVal[wave][j];
    const float2 g = f2[c * 32 + lane];
    SGC_FLUSH_IF_NEW(r);
    ax = fmaf(v, g.x, ax); ay = fmaf(v, g.y, ay);
  }
  // final flush
  if ((unsigned)cur < (unsigned)nNodes) {
    float* dst = out + (size_t)cur * D_FEAT + (lane << 1);
    atomicAdd(dst + 0, ONE_MINUS_ALPHA * ax);
    atomicAdd(dst + 1, ONE_MINUS_ALPHA * ay);
  }
#undef SGC_FLUSH_IF_NEW
}

// ---------------- pass 3: in-place ReLU ----------------
__global__ __launch_bounds__(256) void sgc_relu_kernel(float* __restrict__ out, int n4) {
  int i = blockIdx.x * blockDim.x + threadIdx.x;
  if (i < n4) {
    float4 v = ((float4*)out)[i];
    v.x = fmaxf(v.x, 0.f); v.y = fmaxf(v.y, 0.f);
    v.z = fmaxf(v.z, 0.f); v.w = fmaxf(v.w, 0.f);
    ((float4*)out)[i] = v;
  }
}

extern "C" void kernel_launch(void* const* d_in, const int* in_sizes, int n_in,
                              void* d_out, int out_size, void* d_ws, size_t ws_size,
                              hipStream_t stream) {
  (void)n_in; (void)d_ws; (void)ws_size; (void)out_size;
  const float* features  = (const float*)d_in[0];
  const float* features0 = (const float*)d_in[1];
  const int*   edge_rows = (const int*)d_in[2];
  const int*   edge_cols = (const int*)d_in[3];
  const float* edge_vals = (const float*)d_in[4];
  float* out = (float*)d_out;

  const int nNodesD = in_sizes[0];            // N * 64
  const int nEdges  = in_sizes[2];
  const int nNodes  = nNodesD / D_FEAT;
  const int n4      = nNodesD / 4;            // D=64 -> always divisible

  // Pass 1: out = ALPHA * features0 (also initializes every accumulator row, including those
  // that receive no edges).
  sgc_init_kernel<<<(n4 + 255) / 256, 256, 0, stream>>>(features0, out, n4);

  // Pass 2: segmented scatter-accumulate of (1-ALPHA) * SpMM into out.
  const int nWaves  = (nEdges + EPW - 1) / EPW;
  const int nBlocks = (nWaves + WAVES_PER_BLOCK - 1) / WAVES_PER_BLOCK;
  sgc_scatter_kernel<<<nBlocks, BLOCK_THREADS, 0, stream>>>(
      edge_rows, edge_cols, edge_vals, features, out, nEdges, nNodes);

  // Pass 3: ReLU in place (stream-ordered after all atomics complete).
  sgc_relu_kernel<<<(n4 + 255) / 256, 256, 0, stream>>>(out, n4);
}